// ExperimentNet_31842887533075
// MI455X (gfx1250) — compile-verified
//
#include <hip/hip_runtime.h>

typedef __attribute__((ext_vector_type(2))) float v2f;
typedef __attribute__((ext_vector_type(4))) float v4f;
typedef __attribute__((ext_vector_type(8))) float v8f;

#define HW   16384   // 128*128
#define CCH  256     // C
#define BB   32      // B
#define CR   64      // C / r

// ---------------------------------------------------------------------------
// Kernel 1: m[b][c] = mean(x[b][c][:][:])   — pure HBM-bandwidth streaming.
// One block per contiguous 64 KB (b,c) slice; 256 threads * 16 float4 NT loads.
// ---------------------------------------------------------------------------
__global__ void __launch_bounds__(256)
se_pool_kernel(const float* __restrict__ x, float* __restrict__ m) {
    const int slice = blockIdx.x;                 // flat b*C + c
    const float* p = x + (size_t)slice * HW;
    const int tid = threadIdx.x;

    v4f acc = {0.f, 0.f, 0.f, 0.f};
    #pragma unroll
    for (int i = 0; i < 16; ++i) {
        const v4f* src = (const v4f*)(p + (i * 256 + tid) * 4);
        v4f v = __builtin_nontemporal_load(src);  // stream: don't pollute caches
        acc += v;
    }
    float s = acc.x + acc.y + acc.z + acc.w;

    // wave32 reduction
    #pragma unroll
    for (int off = 16; off > 0; off >>= 1)
        s += __shfl_down(s, off, 32);

    __shared__ float red[8];
    const int wid  = tid >> 5;
    const int lane = tid & 31;
    if (lane == 0) red[wid] = s;
    __syncthreads();
    if (tid == 0) {
        float t = 0.f;
        #pragma unroll
        for (int i = 0; i < 8; ++i) t += red[i];
        m[slice] = t * (1.0f / (float)HW);
    }
}

// ---------------------------------------------------------------------------
// Kernel 2: fused SE-MLP on the 32x256 pooled features, fp32 WMMA 16x16x4.
//   h = relu(m @ W1^T)         (32x64)   stage 1: 1 tile / wave
//   g = sigmoid(h @ W2^T)      (32x256)  stage 2: 4 tiles / wave
//   p = g * m ; out = p @ W3^T + b3      stage 3: VALU
// Single workgroup of 256 threads (8 wave32s).
// ---------------------------------------------------------------------------
__global__ void __launch_bounds__(256)
se_mlp_kernel(const float* __restrict__ m,
              const float* __restrict__ W1,   // (64, 256)
              const float* __restrict__ W2,   // (256, 64)
              const float* __restrict__ W3,   // (2, 256)
              const float* __restrict__ b3,   // (2,)
              float* __restrict__ out) {      // (32, 2)
    __shared__ float h[BB * CR];      // 8 KB
    __shared__ float pbuf[BB * CCH];  // 32 KB

    const int tid  = threadIdx.x;
    const int w    = tid >> 5;     // wave id 0..7
    const int lane = tid & 31;
    const int nl   = lane & 15;    // N (or M-row) within tile
    const int kh   = lane >> 4;    // K-half: lanes 0-15 -> K{0,1}, 16-31 -> K{2,3}
    const int mb   = kh * 8;       // accumulator M base (D layout)

    // ---- stage 1: h = relu(m @ W1^T); wave w -> M-tile (w>>2), N-tile (w&3)
    {
        const int mt = w >> 2, nt = w & 3;
        const int row = mt * 16 + nl;   // A row in m
        const int col = nt * 16 + nl;   // B col == W1 row (W1 is K-contiguous)
        v8f acc = {};
        #pragma unroll 8
        for (int k = 0; k < CCH; k += 4) {
            v2f a = *(const v2f*)(m  + row * CCH + k + kh * 2);
            v2f b = *(const v2f*)(W1 + col * CCH + k + kh * 2);
            acc = __builtin_amdgcn_wmma_f32_16x16x4_f32(
                false, a, false, b, (short)0, acc, false, false);
        }
        #pragma unroll
        for (int i = 0; i < 8; ++i) {
            float v = acc[i];
            h[(mt * 16 + mb + i) * CR + nt * 16 + nl] = v > 0.f ? v : 0.f;
        }
    }
    __syncthreads();

    // ---- stage 2: p = sigmoid(h @ W2^T) * m ; wave w -> N-tiles {2w, 2w+1}
    for (int t = 0; t < 4; ++t) {
        const int mt = t >> 1;
        const int nt = w * 2 + (t & 1);
        const int row = mt * 16 + nl;
        const int col = nt * 16 + nl;   // W2 row (K-contiguous, K=64)
        v8f acc = {};
        #pragma unroll 8
        for (int k = 0; k < CR; k += 4) {
            v2f a = *(const v2f*)(h  + row * CR + k + kh * 2);
            v2f b = *(const v2f*)(W2 + col * CR + k + kh * 2);
            acc = __builtin_amdgcn_wmma_f32_16x16x4_f32(
                false, a, false, b, (short)0, acc, false, false);
        }
        #pragma unroll
        for (int i = 0; i < 8; ++i) {
            const int M = mt * 16 + mb + i;
            const int N = nt * 16 + nl;
            const float g = 1.0f / (1.0f + __expf(-acc[i]));
            pbuf[M * CCH + N] = g * m[M * CCH + N];
        }
    }
    __syncthreads();

    // ---- stage 3: out[b][j] = dot(p[b,:], W3[j,:]) + b3[j]   (32x2)
    if (tid < 64) {
        const int b = tid >> 1, j = tid & 1;
        float s = b3[j];
        #pragma unroll 8
        for (int c = 0; c < CCH; ++c)
            s += pbuf[b * CCH + c] * W3[j * CCH + c];
        out[b * 2 + j] = s;
    }
}

extern "C" void kernel_launch(void* const* d_in, const int* in_sizes, int n_in,
                              void* d_out, int out_size, void* d_ws, size_t ws_size,
                              hipStream_t stream) {
    (void)in_sizes; (void)n_in; (void)out_size; (void)ws_size;
    const float* x  = (const float*)d_in[0];
    const float* W1 = (const float*)d_in[1];
    const float* W2 = (const float*)d_in[2];
    const float* W3 = (const float*)d_in[3];
    const float* b3 = (const float*)d_in[4];

    float* m = (float*)d_ws;   // 32*256 fp32 pooled means (32 KB scratch)

    // Stage A: HBM-bound mean pool, 8192 blocks x 256 threads.
    se_pool_kernel<<<dim3(BB * CCH), dim3(256), 0, stream>>>(x, m);
    // Stage B: tiny fused WMMA MLP, one workgroup.
    se_mlp_kernel<<<dim3(1), dim3(256), 0, stream>>>(m, W1, W2, W3, b3,
                                                     (float*)d_out);
}